// ParamMemSGD_3204045603576
// MI455X (gfx1250) — compile-verified
//
#include <hip/hip_runtime.h>
#include <hip/hip_bf16.h>

typedef float v2f __attribute__((ext_vector_type(2)));
typedef float v8f __attribute__((ext_vector_type(8)));

#define HB   64      // H
#define H2B  128     // 2H
#define LB   2048
#define BB   256
#define NPOS 1023    // scan positions per epoch
#define NSTEP 3069   // 3 * 1023
#define LRC  0.05f

// -------------------------------------------------------------------------
// Kernel 1: e = embed[seq]; ff = relu(e@ffw1^T+b1)@ffw2^T+b2; hs = LN(e+ff)
// 128 threads = 4 waves; each wave owns 16 tokens; WMMA f32 16x16x4.
// -------------------------------------------------------------------------
__global__ __launch_bounds__(128) void k_ffw_ln(
    const int* __restrict__ seq, const float* __restrict__ embed,
    const float* __restrict__ ffw1, const float* __restrict__ ffb1,
    const float* __restrict__ ffw2, const float* __restrict__ ffb2,
    const float* __restrict__ ln_g, const float* __restrict__ ln_b,
    float* __restrict__ hs)
{
    __shared__ float sE[4][16 * 68];     // e tile, stride 68 (conflict-free frags)
    __shared__ float sA[4][16 * 132];    // relu(gemm1) tile, stride 132; reused for hs rows (stride 68)
    __shared__ float sBias[320];         // ffb1[128] | ffb2[64] | ln_g[64] | ln_b[64]

    const int tid  = threadIdx.x;
    const int wave = tid >> 5;
    const int lane = tid & 31;
    const int m    = lane & 15;          // token row within tile / N column index
    const int hi   = lane >> 4;          // K-half selector for fragments

    for (int idx = tid; idx < 320; idx += 128) {
        float v;
        if      (idx < 128) v = ffb1[idx];
        else if (idx < 192) v = ffb2[idx - 128];
        else if (idx < 256) v = ln_g[idx - 192];
        else                v = ln_b[idx - 256];
        sBias[idx] = v;
    }
    __syncthreads();

    const int tokBase = blockIdx.x * 64 + wave * 16;   // 16 tokens per wave
    float* et  = &sE[wave][0];
    float* a1w = &sA[wave][0];

    // ---- load 16 embedding rows: lane (m,hi) loads cols 32*hi..32*hi+31 ----
    {
        const int eidx = seq[tokBase + m];
        const float* erow = embed + (size_t)eidx * HB + hi * 32;
        #pragma unroll
        for (int jj = 0; jj < 8; ++jj) {
            float4 v = *(const float4*)(erow + 4 * jj);
            *(float4*)(et + m * 68 + hi * 32 + 4 * jj) = v;
        }
    }
    __syncthreads();

    // ---- GEMM1: a1[16x128] = relu(e[16x64] @ ffw1^T + ffb1) ----
    #pragma unroll
    for (int nt = 0; nt < 8; ++nt) {
        v8f c = {0.f, 0.f, 0.f, 0.f, 0.f, 0.f, 0.f, 0.f};
        #pragma unroll
        for (int kk = 0; kk < 16; ++kk) {
            v2f a = *(const v2f*)(et + m * 68 + 4 * kk + 2 * hi);
            v2f b = *(const v2f*)(ffw1 + (size_t)(nt * 16 + m) * HB + 4 * kk + 2 * hi);
            c = __builtin_amdgcn_wmma_f32_16x16x4_f32(false, a, false, b,
                                                      (short)0, c, false, false);
        }
        const float bias = sBias[nt * 16 + m];   // ffb1[n0+col], col == m (lane&15)
        #pragma unroll
        for (int r = 0; r < 8; ++r) {
            float v = fmaxf(c[r] + bias, 0.f);
            a1w[(r + 8 * hi) * 132 + nt * 16 + m] = v;
        }
    }
    __syncthreads();

    // ---- GEMM2: ff[16x64] = a1[16x128] @ ffw2^T + ffb2 ----
    v8f acc2[4];
    #pragma unroll
    for (int nt = 0; nt < 4; ++nt)
        acc2[nt] = (v8f){0.f, 0.f, 0.f, 0.f, 0.f, 0.f, 0.f, 0.f};
    #pragma unroll
    for (int nt = 0; nt < 4; ++nt) {
        #pragma unroll
        for (int kk = 0; kk < 32; ++kk) {
            v2f a = *(const v2f*)(a1w + m * 132 + 4 * kk + 2 * hi);
            v2f b = *(const v2f*)(ffw2 + (size_t)(nt * 16 + m) * H2B + 4 * kk + 2 * hi);
            acc2[nt] = __builtin_amdgcn_wmma_f32_16x16x4_f32(false, a, false, b,
                                                             (short)0, acc2[nt], false, false);
        }
    }
    __syncthreads();   // a1 tile now dead; reuse buffer for hs rows (stride 68)

    // ---- residual + bias -> hsrow tile ----
    #pragma unroll
    for (int nt = 0; nt < 4; ++nt) {
        const int c0 = nt * 16 + m;
        const float b2v = sBias[128 + c0];
        #pragma unroll
        for (int r = 0; r < 8; ++r) {
            const int row = r + 8 * hi;
            float v = acc2[nt][r] + b2v + et[row * 68 + c0];
            a1w[row * 68 + c0] = v;
        }
    }
    __syncthreads();

    // ---- LayerNorm per row; lane (m,hi) owns features 32*hi..32*hi+31 ----
    {
        float vals[32];
        float s1 = 0.f, s2 = 0.f;
        #pragma unroll
        for (int j = 0; j < 32; ++j) {
            float x = a1w[m * 68 + 32 * hi + j];
            vals[j] = x;
            s1 += x;
            s2 += x * x;
        }
        s1 += __shfl_xor(s1, 16, 32);
        s2 += __shfl_xor(s2, 16, 32);
        const float mu  = s1 * (1.f / 64.f);
        const float var = s2 * (1.f / 64.f) - mu * mu;
        const float inv = rsqrtf(var + 1e-5f);

        float* orow = hs + (size_t)(tokBase + m) * HB + 32 * hi;
        #pragma unroll
        for (int jj = 0; jj < 8; ++jj) {
            float4 o;
            float* op = (float*)&o;
            #pragma unroll
            for (int u = 0; u < 4; ++u) {
                const int j = 4 * jj + u;
                const int f = 32 * hi + j;
                op[u] = (vals[j] - mu) * inv * sBias[192 + f] + sBias[256 + f];
            }
            *(float4*)(orow + 4 * jj) = o;
        }
    }
}

// -------------------------------------------------------------------------
// Kernel 2: 3069 sequential SGD steps per batch element.
// One block (128 thr = 4 waves) per batch chain; weights register-resident.
// Roles: (i,q) = (t&31, t>>5) for 32x4 slices; (n,half) = (t&63, t>>6) for 64x2.
// q == wave index, so all broadcast LDS reads are wave-uniform (bank-free).
// -------------------------------------------------------------------------
__global__ __launch_bounds__(128) void k_scan(
    const float* __restrict__ hs,
    const float* __restrict__ w1g, const float* __restrict__ b1g,
    const float* __restrict__ w2g, const float* __restrict__ b2g,
    float* __restrict__ ctx)
{
    const int t    = threadIdx.x;
    const int b    = blockIdx.x;
    const int i    = t & 31, q    = t >> 5;
    const int n    = t & 63, half = t >> 6;
    const int col  = t & 63;       // kv element this thread fetches
    const int isv  = t >> 6;       // 0: fetch k[col], 1: fetch v[col]

    __shared__ float kbuf[64], vbuf[64], hbuf[32], dybuf[64], dhbuf[32];
    __shared__ float p1[128], p2[128], p3[128];

    float w1r[16], w2rr[16], w2cr[16];
    #pragma unroll
    for (int jj = 0; jj < 16; ++jj) w1r[jj]  = w1g[i * 64 + 16 * q + jj];
    #pragma unroll
    for (int ii = 0; ii < 16; ++ii) w2rr[ii] = w2g[n * 32 + 16 * half + ii];
    #pragma unroll
    for (int nn = 0; nn < 16; ++nn) w2cr[nn] = w2g[(16 * q + nn) * 32 + i];
    float b1r = (q == 0)    ? b1g[i] : 0.f;
    float b2r = (half == 0) ? b2g[n] : 0.f;

    const float* base = hs + (size_t)b * LB * HB;
    float hp_mask = 0.f;

    // prefetch queue, depth 3 (3069 = 3*1023 -> unroll by 3, no dynamic regs)
    float pf0 = base[(size_t)(0 + isv) * HB + col];   // pos 0
    float pf1 = base[(size_t)(2 + isv) * HB + col];   // pos 1
    float pf2 = base[(size_t)(4 + isv) * HB + col];   // pos 2
    int cpos = 0;   // position (s mod 1023) of the current step

    auto step = [&](float& pfreg, bool dopf) {
        // 1. publish k/v for this step, then prefetch 3 steps ahead
        if (isv) vbuf[col] = pfreg; else kbuf[col] = pfreg;
        __syncthreads();
        if (dopf) {
            int pp = cpos + 3; if (pp >= NPOS) pp -= NPOS;
            pfreg = base[(size_t)(2 * pp + isv) * HB + col];
        }
        // 2. h_pre partials: thread (i,q) covers j in [16q,16q+16)
        float acc = 0.f;
        #pragma unroll
        for (int jj = 0; jj < 16; ++jj) acc = fmaf(w1r[jj], kbuf[16 * q + jj], acc);
        p1[32 * q + i] = acc;
        __syncthreads();
        // 3. h = relu(hpre)
        if (t < 32) {
            float hp = p1[i] + p1[32 + i] + p1[64 + i] + p1[96 + i] + b1r;
            hp_mask = (hp > 0.f) ? 1.f : 0.f;
            hbuf[i] = fmaxf(hp, 0.f);
        }
        __syncthreads();
        // 4. y partials: thread (n,half) covers i in [16half,16half+16)
        acc = 0.f;
        #pragma unroll
        for (int ii = 0; ii < 16; ++ii) acc = fmaf(w2rr[ii], hbuf[16 * half + ii], acc);
        p2[64 * half + n] = acc;
        __syncthreads();
        // 5. dy = 2/H * (y - v);  b2 -= LR*dy
        if (t < 64) {
            float y  = p2[n] + p2[64 + n] + b2r;
            float dy = (y - vbuf[n]) * (2.0f / 64.0f);
            dybuf[n] = dy;
            b2r -= LRC * dy;
        }
        __syncthreads();
        // 6. dh partials: thread (i,q) covers n in [16q,16q+16) via column slice
        acc = 0.f;
        #pragma unroll
        for (int nn = 0; nn < 16; ++nn) acc = fmaf(w2cr[nn], dybuf[16 * q + nn], acc);
        p3[32 * q + i] = acc;
        __syncthreads();
        // 7. dh = (w2^T dy) * relu'(hpre);  b1 -= LR*dh
        if (t < 32) {
            float dh = (p3[i] + p3[32 + i] + p3[64 + i] + p3[96 + i]) * hp_mask;
            dhbuf[i] = dh;
            b1r -= LRC * dh;
        }
        __syncthreads();
        // 8. weight updates (read-only LDS from here)
        {
            const float dh_i = dhbuf[i];
            #pragma unroll
            for (int jj = 0; jj < 16; ++jj)
                w1r[jj] = fmaf(-LRC * dh_i, kbuf[16 * q + jj], w1r[jj]);
            const float hv = hbuf[i];
            #pragma unroll
            for (int nn = 0; nn < 16; ++nn)
                w2cr[nn] = fmaf(-LRC * hv, dybuf[16 * q + nn], w2cr[nn]);
            const float dyn = dybuf[n];
            #pragma unroll
            for (int ii = 0; ii < 16; ++ii)
                w2rr[ii] = fmaf(-LRC * dyn, hbuf[16 * half + ii], w2rr[ii]);
        }
        __syncthreads();
        cpos++; if (cpos == NPOS) cpos = 0;
    };

    #pragma unroll 1
    for (int it = 0; it < NPOS; ++it) {
        const bool dopf = (it < NPOS - 1);
        step(pf0, dopf);
        step(pf1, dopf);
        step(pf2, dopf);
    }

    // ---- final forward with q = hs[b, L-1, :] ----
    if (t < 64) kbuf[t] = base[(size_t)(LB - 1) * HB + t];
    __syncthreads();
    float acc = 0.f;
    #pragma unroll
    for (int jj = 0; jj < 16; ++jj) acc = fmaf(w1r[jj], kbuf[16 * q + jj], acc);
    p1[32 * q + i] = acc;
    __syncthreads();
    if (t < 32) {
        float hp = p1[i] + p1[32 + i] + p1[64 + i] + p1[96 + i] + b1r;
        hbuf[i] = fmaxf(hp, 0.f);
    }
    __syncthreads();
    acc = 0.f;
    #pragma unroll
    for (int ii = 0; ii < 16; ++ii) acc = fmaf(w2rr[ii], hbuf[16 * half + ii], acc);
    p2[64 * half + n] = acc;
    __syncthreads();
    if (t < 64) ctx[(size_t)b * HB + n] = p2[n] + p2[64 + n] + b2r;
}

// -------------------------------------------------------------------------
// Kernel 3: out[256x64] = ctx[256x64] @ out_w^T + out_b, WMMA f32 16x16x4.
// One wave per 16x16 tile; grid = (256/16)*(64/16) = 64 blocks of 32 threads.
// -------------------------------------------------------------------------
__global__ __launch_bounds__(32) void k_out(
    const float* __restrict__ ctx, const float* __restrict__ out_w,
    const float* __restrict__ out_b, float* __restrict__ out)
{
    const int lane = threadIdx.x;
    const int ml   = lane & 15;
    const int hi   = lane >> 4;
    const int m0   = (blockIdx.x >> 2) * 16;
    const int n0   = (blockIdx.x & 3) * 16;

    v8f c = {0.f, 0.f, 0.f, 0.f, 0.f, 0.f, 0.f, 0.f};
    #pragma unroll
    for (int kk = 0; kk < 16; ++kk) {
        v2f a = *(const v2f*)(ctx   + (size_t)(m0 + ml) * HB + 4 * kk + 2 * hi);
        v2f b = *(const v2f*)(out_w + (size_t)(n0 + ml) * HB + 4 * kk + 2 * hi);
        c = __builtin_amdgcn_wmma_f32_16x16x4_f32(false, a, false, b,
                                                  (short)0, c, false, false);
    }
    const float bias = out_b[n0 + ml];
    #pragma unroll
    for (int r = 0; r < 8; ++r)
        out[(size_t)(m0 + r + 8 * hi) * 64 + n0 + ml] = c[r] + bias;
}

// -------------------------------------------------------------------------
extern "C" void kernel_launch(void* const* d_in, const int* in_sizes, int n_in,
                              void* d_out, int out_size, void* d_ws, size_t ws_size,
                              hipStream_t stream) {
    const int*   seq   = (const int*)  d_in[0];
    const float* embed = (const float*)d_in[1];
    const float* ffw1  = (const float*)d_in[2];
    const float* ffb1  = (const float*)d_in[3];
    const float* ffw2  = (const float*)d_in[4];
    const float* ffb2  = (const float*)d_in[5];
    const float* ln_g  = (const float*)d_in[6];
    const float* ln_b  = (const float*)d_in[7];
    const float* w1    = (const float*)d_in[8];
    const float* b1    = (const float*)d_in[9];
    const float* w2    = (const float*)d_in[10];
    const float* b2    = (const float*)d_in[11];
    const float* out_w = (const float*)d_in[12];
    const float* out_b = (const float*)d_in[13];

    float* hs  = (float*)d_ws;                          // 256*2048*64 f32 = 128 MiB
    float* ctx = hs + (size_t)BB * LB * HB;             // 256*64 f32

    k_ffw_ln<<<(BB * LB) / 64, 128, 0, stream>>>(seq, embed, ffw1, ffb1, ffw2, ffb2,
                                                 ln_g, ln_b, hs);
    k_scan<<<BB, 128, 0, stream>>>(hs, w1, b1, w2, b2, ctx);
    k_out<<<64, 32, 0, stream>>>(ctx, out_w, out_b, (float*)d_out);
}